// SpecDecodingPair_70050916598207
// MI455X (gfx1250) — compile-verified
//
#include <hip/hip_runtime.h>
#include <stdint.h>

// Problem constants (match reference)
#define BB 32
#define KK 8
#define VV 50257
#define DD 1024
#define CH 1024                 // elements per streamed chunk
#define NCH 50                  // ceil(VV / CH)
#define SPLIT 2                 // row halves streamed by independent blocks
#define CPS (NCH / SPLIT)       // chunks per block
#define WS_STRIDE 64            // padded chunk-sum stride per row
#define NROW (BB * KK)

#if __has_builtin(__builtin_amdgcn_s_wait_asynccnt)
#define WAIT_ASYNCCNT(n) __builtin_amdgcn_s_wait_asynccnt(n)
#else
#define WAIT_ASYNCCNT(n) asm volatile("s_wait_asynccnt %0" ::"n"(n) : "memory")
#endif

// CDNA5 async global -> LDS copy (32-bit, GV addressing). One instruction per
// wave moves 128B; tracked with ASYNCcnt. The "memory" clobber also stops the
// compiler from assuming the LDS buffers are never written.
__device__ __forceinline__ void async_copy_f32(uint32_t lds_byte_addr, const float* g) {
  asm volatile("global_load_async_to_lds_b32 %0, %1, off"
               :: "v"(lds_byte_addr), "v"((uint64_t)(uintptr_t)g)
               : "memory");
}

// ---------------------------------------------------------------------------
// Pass 1: stream draft/target probs through LDS (double-buffered async copies)
// and emit per-chunk sums of pD and of max(pT - pD, 0).
// ---------------------------------------------------------------------------
__global__ __launch_bounds__(256) void k_stream(const float* __restrict__ dpA,
                                                const float* __restrict__ tpA,
                                                float* __restrict__ wsD,
                                                float* __restrict__ wsC) {
  __shared__ float sD[2][CH];
  __shared__ float sT[2][CH];
  __shared__ float redD[8], redC[8];

  const int row  = blockIdx.x;
  const int c0   = blockIdx.y * CPS;
  const int tid  = threadIdx.x;
  const int lane = tid & 31;
  const int wid  = tid >> 5;
  const float* dp = dpA + (size_t)row * VV;
  const float* tp = tpA + (size_t)row * VV;

  auto issue = [&](int c, int s) {
#pragma unroll
    for (int j = 0; j < 4; ++j) {
      const int li = tid + j * 256;
      const int e  = c * CH + li;
      const int ec = (e < VV) ? e : (VV - 1);   // keep tail loads in-bounds
      async_copy_f32((uint32_t)(uintptr_t)&sD[s][li], dp + ec);
      async_copy_f32((uint32_t)(uintptr_t)&sT[s][li], tp + ec);
    }
  };

  issue(c0, 0);
  for (int i = 0; i < CPS; ++i) {
    const int c = c0 + i;
    if (i + 1 < CPS) {
      issue(c + 1, (i + 1) & 1);   // prefetch next chunk into other buffer
      WAIT_ASYNCCNT(8);            // 8 async ops/wave/chunk -> chunk c done
    } else {
      WAIT_ASYNCCNT(0);
    }
    __syncthreads();               // all waves' copies for chunk c visible

    const int s = i & 1;
    float aD = 0.f, aC = 0.f;
#pragma unroll
    for (int j = 0; j < 4; ++j) {
      const int li = tid + j * 256;
      const int e  = c * CH + li;
      const float d = sD[s][li];
      const float t = sT[s][li];
      if (e < VV) { aD += d; aC += fmaxf(t - d, 0.f); }
    }
#pragma unroll
    for (int o = 16; o > 0; o >>= 1) {
      aD += __shfl_xor(aD, o, 32);
      aC += __shfl_xor(aC, o, 32);
    }
    if (lane == 0) { redD[wid] = aD; redC[wid] = aC; }
    __syncthreads();               // consume done: buffer may be overwritten
    if (tid == 0) {
      float tD = 0.f, tC = 0.f;
#pragma unroll
      for (int w = 0; w < 8; ++w) { tD += redD[w]; tC += redC[w]; }
      wsD[row * WS_STRIDE + c] = tD;
      wsC[row * WS_STRIDE + c] = tC;
    }
  }
}

// ---------------------------------------------------------------------------
// Pass 2 helper: exact in-chunk inverse-CDF count via wave32 shuffle scan.
// Counts positions i in chunk `cstar` with (run + cumsum(i)) < u.
// ---------------------------------------------------------------------------
template <typename F>
__device__ __forceinline__ int count_in_chunk(int cstar, float run, float u,
                                              int lane, F getv) {
  int cnt = 0;
  for (int t = 0; t < CH / 32; ++t) {
    const int e = cstar * CH + t * 32 + lane;
    float incl = (e < VV) ? getv(e) : 0.f;
#pragma unroll
    for (int o = 1; o < 32; o <<= 1) {
      const float x = __shfl_up(incl, o, 32);
      if (lane >= o) incl += x;
    }
    const bool pred = (e < VV) && ((run + incl) < u);
    cnt += (int)__popcll(__ballot(pred));
    run += __shfl(incl, 31, 32);
  }
  return cnt;
}

// One wave per (b,k) row: draft sample, accept ratio, residual resample.
// Chunk re-reads hit L2 (103 MB working set < 192 MB L2).
__global__ __launch_bounds__(32) void k_sample(const float* __restrict__ dpA,
                                               const float* __restrict__ tpA,
                                               const float* __restrict__ wsD,
                                               const float* __restrict__ wsC,
                                               const float* __restrict__ u_sample,
                                               const float* __restrict__ u_accept,
                                               const float* __restrict__ u_resample,
                                               int* __restrict__ res_ids,
                                               int* __restrict__ accept_flag) {
  const int row  = blockIdx.x;
  const int lane = threadIdx.x;
  const float* dp = dpA + (size_t)row * VV;
  const float* tp = tpA + (size_t)row * VV;
  const float us = u_sample[row], ua = u_accept[row], ur = u_resample[row];

  // --- draft token: idx = #{ i : cumsum(pD)_i < u_sample } ---
  float run = 0.f; int cstar = -1; int base = 0;
  for (int c = 0; c < NCH; ++c) {
    const float s = wsD[row * WS_STRIDE + c];
    if (run + s >= us) { cstar = c; break; }
    run += s; base += CH;
  }
  int idx;
  if (cstar < 0) idx = VV - 1;
  else {
    const int cnt = count_in_chunk(cstar, run, us, lane,
                                   [&](int e) { return dp[e]; });
    idx = base + cnt; if (idx > VV - 1) idx = VV - 1;
  }
  const float pD = dp[idx];
  const float pT = tp[idx];
  const float ratio = fminf(pT / pD, 1.0f);

  // --- resample from normalized max(0, pT - pD); compare cumsum < u*S ---
  float S = 0.f;
  for (int c = 0; c < NCH; ++c) S += wsC[row * WS_STRIDE + c];
  int rid;
  if (!(S > 0.f)) {
    rid = (ur > 0.f) ? (VV - 1) : 0;   // zero residual -> cdf==0 everywhere
  } else {
    const float uu = ur * S;
    run = 0.f; cstar = -1; base = 0;
    for (int c = 0; c < NCH; ++c) {
      const float s = wsC[row * WS_STRIDE + c];
      if (run + s >= uu) { cstar = c; break; }
      run += s; base += CH;
    }
    if (cstar < 0) rid = VV - 1;
    else {
      const int cnt = count_in_chunk(cstar, run, uu, lane,
                                     [&](int e) { return fmaxf(tp[e] - dp[e], 0.f); });
      rid = base + cnt; if (rid > VV - 1) rid = VV - 1;
    }
  }
  if (lane == 0) {
    res_ids[row]     = rid;
    accept_flag[row] = (ua < ratio) ? 1 : 0;
  }
}

// accepted_count = leading k's where ALL batch rows accept.
__global__ void k_count(const int* __restrict__ accept_flag,
                        int* __restrict__ ws_count,
                        float* __restrict__ out_count) {
  if (threadIdx.x == 0 && blockIdx.x == 0) {
    int cnt = 0;
    for (int k = 0; k < KK; ++k) {
      int all = 1;
      for (int b = 0; b < BB; ++b) all &= accept_flag[b * KK + k];
      if (!all) break;
      ++cnt;
    }
    *ws_count  = cnt;
    *out_count = (float)cnt;
  }
}

// out[b,k,:] = (k < count) ? draft_sequence[b,k,:] : lm_head[res_id[b,k],:]
__global__ __launch_bounds__(256) void k_assemble(const float* __restrict__ draft_seq,
                                                  const float* __restrict__ lm_head,
                                                  const int* __restrict__ res_ids,
                                                  const int* __restrict__ ws_count,
                                                  float* __restrict__ out) {
  const int row = blockIdx.x;
  const int k   = row & (KK - 1);
  const int cnt = *ws_count;
  const float* src = (k < cnt) ? (draft_seq + (size_t)row * DD)
                               : (lm_head + (size_t)res_ids[row] * DD);
  const float4 v = ((const float4*)src)[threadIdx.x];
  ((float4*)(out + (size_t)row * DD))[threadIdx.x] = v;
}

extern "C" void kernel_launch(void* const* d_in, const int* in_sizes, int n_in,
                              void* d_out, int out_size, void* d_ws, size_t ws_size,
                              hipStream_t stream) {
  (void)in_sizes; (void)n_in; (void)out_size; (void)ws_size;
  const float* draft_probs  = (const float*)d_in[0];
  const float* target_probs = (const float*)d_in[1];
  const float* draft_seq    = (const float*)d_in[2];
  const float* lm_head      = (const float*)d_in[3];
  const float* u_sample     = (const float*)d_in[4];
  const float* u_accept     = (const float*)d_in[5];
  const float* u_resample   = (const float*)d_in[6];
  float* out = (float*)d_out;

  // Workspace carve-up (~133 KB total)
  float* wsD       = (float*)d_ws;                 // [NROW][WS_STRIDE]
  float* wsC       = wsD + NROW * WS_STRIDE;       // [NROW][WS_STRIDE]
  int*   res_ids   = (int*)(wsC + NROW * WS_STRIDE);
  int*   accept_fl = res_ids + NROW;
  int*   ws_count  = accept_fl + NROW;

  k_stream<<<dim3(NROW, SPLIT), 256, 0, stream>>>(draft_probs, target_probs, wsD, wsC);
  k_sample<<<NROW, 32, 0, stream>>>(draft_probs, target_probs, wsD, wsC,
                                    u_sample, u_accept, u_resample,
                                    res_ids, accept_fl);
  k_count<<<1, 32, 0, stream>>>(accept_fl, ws_count, out + (size_t)NROW * DD);
  k_assemble<<<NROW, 256, 0, stream>>>(draft_seq, lm_head, res_ids, ws_count, out);
}